// BaseTransform_40690520163038
// MI455X (gfx1250) — compile-verified
//
#include <hip/hip_runtime.h>

// ---------------- config (matches reference) ----------------
#define BB   4
#define NN   1
#define DD   118
#define FH   32
#define FW   88
#define CC   80
#define NXX  360
#define NYY  360
#define YX   (NXX * NYY)                   // 129600
#define NPTS (BB * NN * DD * FH * FW)      // 1,329,152
#define PT_PER_B (NN * DD * FH * FW)       // 332,288
#define J_PER_PT (CC / 4)                  // 20 float4 chunks per point

typedef __attribute__((ext_vector_type(2))) float v2f;
typedef __attribute__((ext_vector_type(8))) float v8f;

// ---------------- 3x3 inverse via adjugate ----------------
__device__ inline void inv3(const float* m, float* o) {
    float a = m[0], b = m[1], c = m[2];
    float d = m[3], e = m[4], f = m[5];
    float g = m[6], h = m[7], i = m[8];
    float A =  (e * i - f * h);
    float Bc = -(d * i - f * g);
    float Cc =  (d * h - e * g);
    float det = a * A + b * Bc + c * Cc;
    float r = 1.0f / det;
    o[0] =  A * r;              o[1] = -(b * i - c * h) * r;  o[2] =  (b * f - c * e) * r;
    o[3] =  Bc * r;             o[4] =  (a * i - c * g) * r;  o[5] = -(a * f - c * d) * r;
    o[6] =  Cc * r;             o[7] = -(a * h - b * g) * r;  o[8] =  (a * e - b * d) * r;
}

// ---------------- block-diagonal 16x16 matmul via 4x V_WMMA_F32_16X16X4_F32
// Lf/Rt/Dst are 16x16 row-major LDS tiles. All 32 lanes participate (EXEC all 1s).
// A 16x4 slice layout (ISA 7.12.2): lanes 0-15 hold rows, v0=K(4s+0), v1=K(4s+1);
// lanes 16-31: v0=K(4s+2), v1=K(4s+3).
// B 4x16 slice: lanes 0-15 N=lane, v0=row(4s+0), v1=row(4s+1); lanes 16-31: rows 4s+2/4s+3.
// D 16x16: VGPR r, lanes 0-15 -> (M=r, N=lane); lanes 16-31 -> (M=r+8, N=lane-16).
__device__ inline void wmma_mul16(const float* Lf, const float* Rt, float* Dst, int lane) {
    const int half = lane >> 4;
    const int lr   = lane & 15;
    v8f acc = {0.f, 0.f, 0.f, 0.f, 0.f, 0.f, 0.f, 0.f};
#pragma unroll
    for (int s = 0; s < 4; ++s) {
        const int k0 = 4 * s + 2 * half;
        v2f a, b;
        a.x = Lf[lr * 16 + k0];
        a.y = Lf[lr * 16 + k0 + 1];
        b.x = Rt[(k0)     * 16 + lr];
        b.y = Rt[(k0 + 1) * 16 + lr];
        acc = __builtin_amdgcn_wmma_f32_16x16x4_f32(
            /*neg_a=*/false, a, /*neg_b=*/false, b,
            /*c_mod=*/(short)0, acc, /*reuse_a=*/false, /*reuse_b=*/false);
    }
#pragma unroll
    for (int r = 0; r < 8; ++r)
        Dst[(r + 8 * half) * 16 + lr] = acc[r];
}

// ---------------- kernel 1: compose per-(b,n) transforms (runs once, 1 wave)
// ws layout per (b,n): [0..8]=A, [9..11]=cvec, [12..20]=Pinv, [21..23]=post_trans
__global__ __launch_bounds__(32) void bev_xform_kernel(
    const float* __restrict__ rots, const float* __restrict__ trans,
    const float* __restrict__ intrins, const float* __restrict__ post_rots,
    const float* __restrict__ post_trans, const float* __restrict__ l2c_rots,
    const float* __restrict__ l2c_trans, const float* __restrict__ extra_rots,
    const float* __restrict__ extra_trans, float* __restrict__ xf) {
    __shared__ float Lb[256], Rb[256], Db[256];
    const int lane = threadIdx.x;

    for (int i = lane; i < 256; i += 32) { Lb[i] = 0.f; Rb[i] = 0.f; Db[i] = 0.f; }
    __syncthreads();

    // stage operands: Lb = blockdiag(rots), Rb = blockdiag(inv(intrins))
    if (lane < BB * NN) {
        const int bn = lane;
        float Ii[9];
        inv3(intrins + bn * 9, Ii);
#pragma unroll
        for (int r = 0; r < 3; ++r)
#pragma unroll
            for (int c = 0; c < 3; ++c) {
                Lb[(4 * bn + r) * 16 + 4 * bn + c] = rots[bn * 9 + r * 3 + c];
                Rb[(4 * bn + r) * 16 + 4 * bn + c] = Ii[r * 3 + c];
            }
    }
    __syncthreads();
    wmma_mul16(Lb, Rb, Db, lane);          // Db = rots @ inv(intrins)
    __syncthreads();

    if (lane < BB * NN) {                  // Lb = blockdiag(inv(l2c_rots))
        float Li[9];
        inv3(l2c_rots + lane * 9, Li);
#pragma unroll
        for (int r = 0; r < 3; ++r)
#pragma unroll
            for (int c = 0; c < 3; ++c)
                Lb[(4 * lane + r) * 16 + 4 * lane + c] = Li[r * 3 + c];
    }
    __syncthreads();
    wmma_mul16(Lb, Db, Rb, lane);          // Rb = inv(l2c) @ rots @ inv(intrins)
    __syncthreads();

    if (lane < BB * NN) {                  // Lb = blockdiag(extra_rots)
#pragma unroll
        for (int r = 0; r < 3; ++r)
#pragma unroll
            for (int c = 0; c < 3; ++c)
                Lb[(4 * lane + r) * 16 + 4 * lane + c] = extra_rots[lane * 9 + r * 3 + c];
    }
    __syncthreads();
    wmma_mul16(Lb, Rb, Db, lane);          // Db = A (full composed linear part)
    __syncthreads();

    if (lane < BB * NN) {
        const int bn = lane, b = lane;     // N == 1
        float* o = xf + bn * 24;
#pragma unroll
        for (int r = 0; r < 3; ++r)
#pragma unroll
            for (int c = 0; c < 3; ++c)
                o[r * 3 + c] = Db[(4 * bn + r) * 16 + 4 * bn + c];
        // cvec = E @ (Linv @ (trans - l2c_trans)) + extra_trans
        float Li[9];
        inv3(l2c_rots + b * 9, Li);
        float t1[3], t2[3];
#pragma unroll
        for (int k = 0; k < 3; ++k) t1[k] = trans[bn * 3 + k] - l2c_trans[b * 3 + k];
#pragma unroll
        for (int k = 0; k < 3; ++k)
            t2[k] = Li[k * 3 + 0] * t1[0] + Li[k * 3 + 1] * t1[1] + Li[k * 3 + 2] * t1[2];
#pragma unroll
        for (int k = 0; k < 3; ++k)
            o[9 + k] = extra_rots[b * 9 + k * 3 + 0] * t2[0] +
                       extra_rots[b * 9 + k * 3 + 1] * t2[1] +
                       extra_rots[b * 9 + k * 3 + 2] * t2[2] + extra_trans[b * 3 + k];
        float Pi[9];
        inv3(post_rots + bn * 9, Pi);
#pragma unroll
        for (int k = 0; k < 9; ++k) o[12 + k] = Pi[k];
#pragma unroll
        for (int k = 0; k < 3; ++k) o[21 + k] = post_trans[bn * 3 + k];
    }
}

// ---------------- kernel 2: zero the output grid ----------------
__global__ __launch_bounds__(256) void bev_zero_kernel(float4* __restrict__ out, int n4) {
    int i = blockIdx.x * 256 + threadIdx.x;
    if (i < n4) out[i] = make_float4(0.f, 0.f, 0.f, 0.f);
}

// ---------------- kernel 3: voxelize + scatter-add (memory bound) ----------
// 20 threads per point (coalesced float4 feature reads); atomics land in L2
// (output grid fits in the 192MB L2).
__global__ __launch_bounds__(256) void bev_scatter_kernel(
    const float* __restrict__ x, const float* __restrict__ xf,
    float* __restrict__ out) {
    const int gid = blockIdx.x * 256 + threadIdx.x;
    const int TOT = NPTS * J_PER_PT;
    if (gid >= TOT) return;
    const int pt = gid / J_PER_PT;
    const int j  = gid % J_PER_PT;

    const int b  = pt / PT_PER_B;
    const int r0 = pt % PT_PER_B;
    const int di = r0 / (FH * FW);
    const int r1 = r0 % (FH * FW);
    const int h  = r1 / FW;
    const int w  = r1 % FW;

    const float* T = xf + b * 24;          // N == 1

    // frustum point
    const float fx = (float)w * (703.0f / 87.0f);
    const float fy = (float)h * (255.0f / 31.0f);
    const float fz = 1.0f + 0.5f * (float)di;
    // undo image aug: q = Pinv @ (f - post_trans)
    const float px = fx - T[21], py = fy - T[22], pz = fz - T[23];
    const float qx = T[12] * px + T[13] * py + T[14] * pz;
    const float qy = T[15] * px + T[16] * py + T[17] * pz;
    const float qz = T[18] * px + T[19] * py + T[20] * pz;
    // (u*d, v*d, d), then composed affine
    const float sx = qx * qz, sy = qy * qz, sz = qz;
    const float ox = T[0] * sx + T[1] * sy + T[2] * sz + T[9];
    const float oy = T[3] * sx + T[4] * sy + T[5] * sz + T[10];
    const float oz = T[6] * sx + T[7] * sy + T[8] * sz + T[11];

    // voxel coords; match reference truncation (astype(int32) truncates toward 0,
    // so (-1,0) -> voxel 0 and is kept) and true division by DX.
    const float tx = (ox + 54.0f) / 0.3f;
    const float ty = (oy + 54.0f) / 0.3f;
    const float tz = (oz + 10.0f) / 20.0f;
    if (!(tx > -1.0f && tx < 360.0f &&
          ty > -1.0f && ty < 360.0f &&
          tz > -1.0f && tz <   1.0f))
        return;                            // rejected: feature bytes never touched
    const int cx = (int)tx;
    const int cy = (int)ty;

    const float4 f4 = *(const float4*)(x + (size_t)pt * CC + j * 4);
    float* base = out + ((size_t)b * CC + (size_t)j * 4) * YX + (size_t)cy * NXX + cx;
    atomicAdd(base,              f4.x);    // -> global_atomic_add_f32 (no-return)
    atomicAdd(base + 1 * YX,     f4.y);
    atomicAdd(base + 2 * YX,     f4.z);
    atomicAdd(base + 3 * YX,     f4.w);
}

extern "C" void kernel_launch(void* const* d_in, const int* in_sizes, int n_in,
                              void* d_out, int out_size, void* d_ws, size_t ws_size,
                              hipStream_t stream) {
    const float* x          = (const float*)d_in[0];
    const float* rots       = (const float*)d_in[1];
    const float* trans      = (const float*)d_in[2];
    const float* intrins    = (const float*)d_in[3];
    const float* post_rots  = (const float*)d_in[4];
    const float* post_trans = (const float*)d_in[5];
    const float* l2c_rots   = (const float*)d_in[6];
    const float* l2c_trans  = (const float*)d_in[7];
    const float* extra_rots = (const float*)d_in[8];
    const float* extra_trans= (const float*)d_in[9];
    float* out = (float*)d_out;
    float* xf  = (float*)d_ws;             // needs 4*24 floats = 384 bytes

    bev_xform_kernel<<<1, 32, 0, stream>>>(rots, trans, intrins, post_rots, post_trans,
                                           l2c_rots, l2c_trans, extra_rots, extra_trans, xf);

    const int n4 = out_size / 4;           // 10,368,000 float4
    bev_zero_kernel<<<(n4 + 255) / 256, 256, 0, stream>>>((float4*)out, n4);

    const int TOT = NPTS * J_PER_PT;       // 26,583,040 threads
    bev_scatter_kernel<<<(TOT + 255) / 256, 256, 0, stream>>>(x, xf, out);
}